// DoubleAffineNet_62491774156929
// MI455X (gfx1250) — compile-verified
//
#include <hip/hip_runtime.h>

typedef __attribute__((ext_vector_type(2))) float v2f;
typedef __attribute__((ext_vector_type(8))) float v8f;

#define HH 1024
#define WW 1024
#define BB 8
#define NPIX (HH * WW)
#define NBLK 256
#define THREADS 256

// ---------------------------------------------------------------------------
// Stage 1: per-image partial sums of x (images 0..7) and y (images 8..15).
// float4 vectorized, grid-stride, deterministic block partials (no atomics).
// ---------------------------------------------------------------------------
__global__ void dan_reduce_stage1(const float* __restrict__ x,
                                  const float* __restrict__ y,
                                  float* __restrict__ partials) {
  __shared__ float sdata[THREADS];
  const int img = blockIdx.y;  // 0..15
  const float* src = (img < BB) ? (x + (size_t)img * NPIX)
                                : (y + (size_t)(img - BB) * NPIX);
  const float4* src4 = (const float4*)src;
  const int n4 = NPIX / 4;
  float s = 0.f;
  for (int i = blockIdx.x * THREADS + threadIdx.x; i < n4; i += NBLK * THREADS) {
    float4 v = src4[i];
    s += (v.x + v.y) + (v.z + v.w);
  }
  sdata[threadIdx.x] = s;
  __syncthreads();
  for (int off = THREADS / 2; off > 0; off >>= 1) {
    if (threadIdx.x < off) sdata[threadIdx.x] += sdata[threadIdx.x + off];
    __syncthreads();
  }
  if (threadIdx.x == 0) partials[img * NBLK + blockIdx.x] = sdata[0];
}

// ---------------------------------------------------------------------------
// Stage 2: reduce NBLK partials per row, scale -> mean.
// ---------------------------------------------------------------------------
__global__ void dan_reduce_stage2(const float* __restrict__ partials,
                                  float* __restrict__ dst, float scale) {
  __shared__ float sdata[NBLK];
  const int row = blockIdx.x;
  sdata[threadIdx.x] = partials[row * NBLK + threadIdx.x];
  __syncthreads();
  for (int off = NBLK / 2; off > 0; off >>= 1) {
    if (threadIdx.x < off) sdata[threadIdx.x] += sdata[threadIdx.x + off];
    __syncthreads();
  }
  if (threadIdx.x == 0) dst[row] = sdata[0] * scale;
}

// ---------------------------------------------------------------------------
// Single-wave WMMA helper: D(16x16) = A(16x4) * B(4x16), f32.
// A: feat (8x2, rows=batch), B: W (2x9). Lane layout per CDNA5 ISA 7.12.2.
// Returns v8f: lane n (0..15) holds column n for batch rows 0..7 in d[0..7].
// ---------------------------------------------------------------------------
__device__ inline v8f dan_feat_gemm(float featA, float featB, bool rowValid,
                                    const float* __restrict__ Wl, int lane) {
  const int mn = lane & 15;
  const int kg = lane >> 4;  // 0: K=0,1 ; 1: K=2,3 (zero padding)
  v2f a, b;
  a.x = (kg == 0 && rowValid) ? featA : 0.f;  // A[m][0]
  a.y = (kg == 0 && rowValid) ? featB : 0.f;  // A[m][1]
  const bool colValid = (kg == 0) && (mn < 9);
  b.x = colValid ? Wl[mn] : 0.f;       // B[0][n] = W[0][n]
  b.y = colValid ? Wl[9 + mn] : 0.f;   // B[1][n] = W[1][n]
  v8f c = {};
  return __builtin_amdgcn_wmma_f32_16x16x4_f32(
      false, a, false, b, (short)0, c, false, false);
}

// ---------------------------------------------------------------------------
// phi = I + 0.01*(feat @ Wpsi + bpsi); also phi_inv (3x3 adjugate inverse).
// One wave (32 lanes), EXEC all 1s at the WMMA.
// ---------------------------------------------------------------------------
__global__ void dan_affine_phi(const float* __restrict__ means,   // 16
                               const float* __restrict__ Wl,      // 2x9
                               const float* __restrict__ bl,      // 9
                               float* __restrict__ phi_out,       // 72
                               float* __restrict__ phi_inv_out) { // 72
  __shared__ float phi_s[BB * 9];
  const int lane = threadIdx.x;
  const int mn = lane & 15;
  const bool rowValid = mn < BB;
  const float fa = rowValid ? means[mn] : 0.f;        // mean_x[b]
  const float fb = rowValid ? means[BB + mn] : 0.f;   // mean_y[b]
  v8f d = dan_feat_gemm(fa, fb, rowValid, Wl, lane);

  if ((lane >> 4) == 0 && mn < 9) {
    const int n = mn;
    const float eye = (n == 0 || n == 4 || n == 8) ? 1.f : 0.f;
    const float bias = bl[n];
#pragma unroll
    for (int bt = 0; bt < BB; ++bt) {
      const float v = eye + 0.01f * (d[bt] + bias);
      phi_s[bt * 9 + n] = v;
      phi_out[bt * 9 + n] = v;
    }
  }
  __syncthreads();
  if (lane < BB) {
    const float* M = &phi_s[lane * 9];
    const float A = M[0], B = M[1], C = M[2];
    const float D = M[3], E = M[4], F = M[5];
    const float G = M[6], H = M[7], I = M[8];
    const float c0 = E * I - F * H, c1 = F * G - D * I, c2 = D * H - E * G;
    const float id = 1.f / (A * c0 + B * c1 + C * c2);
    float* o = &phi_inv_out[lane * 9];
    o[0] = c0 * id;            o[1] = (C * H - B * I) * id; o[2] = (B * F - C * E) * id;
    o[3] = c1 * id;            o[4] = (A * I - C * G) * id; o[5] = (C * D - A * F) * id;
    o[6] = c2 * id;            o[7] = (B * G - A * H) * id; o[8] = (A * E - B * D) * id;
  }
}

// ---------------------------------------------------------------------------
// Fused warp + reduction: sum of bilinear samples of y at phi_inv*(r,c,1).
// Never materializes y_comp (only its mean is consumed downstream).
// ---------------------------------------------------------------------------
__device__ inline float dan_tap(const float* __restrict__ img, int xi, int yi,
                                float w) {
  const bool valid = (xi >= 0) & (xi < HH) & (yi >= 0) & (yi < WW);
  const int xc = min(max(xi, 0), HH - 1);
  const int yc = min(max(yi, 0), WW - 1);
  return img[xc * WW + yc] * (valid ? w : 0.f);
}

__global__ void dan_warp_stage1(const float* __restrict__ y,
                                const float* __restrict__ phi_inv,  // 72
                                float* __restrict__ partials) {
  __shared__ float sdata[THREADS];
  __shared__ float pi[9];
  const int bt = blockIdx.y;  // batch
  if (threadIdx.x < 9) pi[threadIdx.x] = phi_inv[bt * 9 + threadIdx.x];
  __syncthreads();
  const float p0 = pi[0], p1 = pi[1], p2 = pi[2];
  const float p3 = pi[3], p4 = pi[4], p5 = pi[5];
  const float* img = y + (size_t)bt * NPIX;
  const float invH = 1.f / (HH - 1), invW = 1.f / (WW - 1);
  float s = 0.f;
  for (int p = blockIdx.x * THREADS + threadIdx.x; p < NPIX;
       p += NBLK * THREADS) {
    const int h = p >> 10;
    const int w = p & (WW - 1);
    const float r = h * invH, cc = w * invW;
    const float px = (p0 * r + p1 * cc + p2) * (float)(HH - 1);
    const float py = (p3 * r + p4 * cc + p5) * (float)(WW - 1);
    const float fx = floorf(px), fy = floorf(py);
    const int x0 = (int)fx, y0 = (int)fy;
    const float wx = px - fx, wy = py - fy;
    s += dan_tap(img, x0,     y0,     (1.f - wx) * (1.f - wy));
    s += dan_tap(img, x0,     y0 + 1, (1.f - wx) * wy);
    s += dan_tap(img, x0 + 1, y0,     wx * (1.f - wy));
    s += dan_tap(img, x0 + 1, y0 + 1, wx * wy);
  }
  sdata[threadIdx.x] = s;
  __syncthreads();
  for (int off = THREADS / 2; off > 0; off >>= 1) {
    if (threadIdx.x < off) sdata[threadIdx.x] += sdata[threadIdx.x + off];
    __syncthreads();
  }
  if (threadIdx.x == 0) partials[bt * NBLK + blockIdx.x] = sdata[0];
}

// ---------------------------------------------------------------------------
// psi via WMMA; out = phi + psi - I = phi + 0.01*(feat2 @ Wphi + bphi).
// ---------------------------------------------------------------------------
__global__ void dan_affine_psi_out(const float* __restrict__ means,    // 16
                                   const float* __restrict__ mean_yc,  // 8
                                   const float* __restrict__ Wl,
                                   const float* __restrict__ bl,
                                   const float* __restrict__ phi,      // 72
                                   float* __restrict__ out) {          // 72
  const int lane = threadIdx.x;
  const int mn = lane & 15;
  const bool rowValid = mn < BB;
  const float fa = rowValid ? means[mn] : 0.f;     // mean_x[b]
  const float fb = rowValid ? mean_yc[mn] : 0.f;   // mean_ycomp[b]
  v8f d = dan_feat_gemm(fa, fb, rowValid, Wl, lane);
  if ((lane >> 4) == 0 && mn < 9) {
    const int n = mn;
    const float bias = bl[n];
#pragma unroll
    for (int bt = 0; bt < BB; ++bt)
      out[bt * 9 + n] = phi[bt * 9 + n] + 0.01f * (d[bt] + bias);
  }
}

// ---------------------------------------------------------------------------
extern "C" void kernel_launch(void* const* d_in, const int* in_sizes, int n_in,
                              void* d_out, int out_size, void* d_ws,
                              size_t ws_size, hipStream_t stream) {
  const float* x    = (const float*)d_in[0];
  const float* y    = (const float*)d_in[1];
  const float* Wpsi = (const float*)d_in[2];
  const float* bpsi = (const float*)d_in[3];
  const float* Wphi = (const float*)d_in[4];
  const float* bphi = (const float*)d_in[5];
  float* out = (float*)d_out;

  float* ws       = (float*)d_ws;
  float* means    = ws;            // [0,16)   mean_x[8], mean_y[8]
  float* mean_yc  = ws + 16;       // [16,24)  mean of warped y
  float* phi      = ws + 24;       // [24,96)
  float* phi_inv  = ws + 96;       // [96,168)
  float* partA    = ws + 256;      // 16*NBLK
  float* partW    = ws + 256 + 16 * NBLK;  // 8*NBLK

  const float inv_npix = 1.f / (float)NPIX;

  dim3 gA(NBLK, 16);
  dan_reduce_stage1<<<gA, THREADS, 0, stream>>>(x, y, partA);
  dan_reduce_stage2<<<16, NBLK, 0, stream>>>(partA, means, inv_npix);
  dan_affine_phi<<<1, 32, 0, stream>>>(means, Wpsi, bpsi, phi, phi_inv);
  dim3 gW(NBLK, BB);
  dan_warp_stage1<<<gW, THREADS, 0, stream>>>(y, phi_inv, partW);
  dan_reduce_stage2<<<BB, NBLK, 0, stream>>>(partW, mean_yc, inv_npix);
  dan_affine_psi_out<<<1, 32, 0, stream>>>(means, mean_yc, Wphi, bphi, phi, out);
}